// DifferentiableLengthRegulator_63136019251668
// MI455X (gfx1250) — compile-verified
//
#include <hip/hip_runtime.h>

#define EPSF 1e-8f
#define B_   16
#define C_   256
#define T_   512
#define L_   4096
#define KCH      128   // K chunk resident in LDS
#define ESTRIDE  132   // padded row stride (words); 132 % 64 == 4 -> conflict-free b64

typedef float v2f __attribute__((ext_vector_type(2)));
typedef float v8f __attribute__((ext_vector_type(8)));

// ---------------------------------------------------------------------------
// Kernel A: per-batch inclusive scan of durations -> center & gaussian coef
// center[b,t] = cumsum(w)[t] - 0.5*w[t] ; coef[b,t] = -0.5 / ((w*ss)^2 + eps)
// ---------------------------------------------------------------------------
__global__ __launch_bounds__(T_) void dlr_scan_kernel(const float* __restrict__ w,
                                                      const float* __restrict__ sigma_scale,
                                                      float* __restrict__ center,
                                                      float* __restrict__ coef) {
    __shared__ float s[T_];
    const int b = blockIdx.x;
    const int t = threadIdx.x;
    const float wv = w[b * T_ + t];
    s[t] = wv;
    __syncthreads();
    #pragma unroll
    for (int off = 1; off < T_; off <<= 1) {
        float add = (t >= off) ? s[t - off] : 0.0f;
        __syncthreads();
        s[t] += add;
        __syncthreads();
    }
    const float ss  = sigma_scale[0];
    const float sig = wv * ss;
    center[b * T_ + t] = s[t] - 0.5f * wv;
    coef[b * T_ + t]   = -0.5f / (sig * sig + EPSF);
}

// ---------------------------------------------------------------------------
// Kernel B: fused gaussian-weight generation (LDS) + f32 WMMA GEMM
// One block: batch b, 64-wide column strip. 256 threads = 8 waves.
// Wave w computes rows [32w, 32w+32) x 64 cols via 2x4 tiles of
// v_wmma_f32_16x16x4_f32, K looped in 128-wide LDS-resident chunks.
// ---------------------------------------------------------------------------
__global__ __launch_bounds__(256) void dlr_upsample_kernel(
    const float* __restrict__ x,       // (B, C, T)
    const float* __restrict__ xmask,   // (B, 1, T)
    const float* __restrict__ ymask,   // (B, 1, L)
    const float* __restrict__ center,  // (B, T)
    const float* __restrict__ coef,    // (B, T)
    float* __restrict__ out)           // (B, C, L)
{
    __shared__ float Elds[64 * ESTRIDE];  // masked exp values, [n][k] padded
    __shared__ float s_ctr[T_];
    __shared__ float s_cf[T_];
    __shared__ float s_m[T_];
    __shared__ float s_part[256];
    __shared__ float s_inv[64];

    const int tid   = threadIdx.x;
    const int b     = blockIdx.x >> 6;     // 64 strips per batch
    const int strip = blockIdx.x & 63;
    const int lBase = strip * 64;

    // stage per-t batch data once
    for (int i = tid; i < T_; i += 256) {
        s_ctr[i] = center[b * T_ + i];
        s_cf[i]  = coef[b * T_ + i];
        s_m[i]   = xmask[b * T_ + i];
    }
    __syncthreads();

    // E-generation assignment: column n, k-segment of 32 per chunk
    const int n    = tid & 63;
    const int kseg = tid >> 6;             // 0..3
    const float lpos = (float)(lBase + n);

    // WMMA assignment
    const int lane   = tid & 31;
    const int wv     = tid >> 5;           // 0..7
    const int lane16 = lane & 15;
    const int hi     = lane >> 4;          // 0 or 1 -> K offset +2 (A/B halves)
    const int cBase  = wv * 32;

    const float* xb = x + (size_t)b * C_ * T_;

    v8f acc[2][4];
    #pragma unroll
    for (int mt = 0; mt < 2; ++mt)
        #pragma unroll
        for (int nt = 0; nt < 4; ++nt)
            acc[mt][nt] = v8f{0.f, 0.f, 0.f, 0.f, 0.f, 0.f, 0.f, 0.f};

    float psum = 0.0f;  // unmasked column sum (normalizer)

    for (int kb = 0; kb < T_; kb += KCH) {
        // ---- Phase 1: compute E chunk into LDS (all 256 threads) ----
        {
            const int kb0 = kseg * 32;
            float* erow = &Elds[n * ESTRIDE + kb0];
            #pragma unroll 8
            for (int i = 0; i < 32; ++i) {
                const int t  = kb + kb0 + i;
                const float mu = lpos - s_ctr[t];
                const float e  = __expf(s_cf[t] * mu * mu);
                psum += e;                 // normalizer uses UNmasked weights
                erow[i] = e * s_m[t];      // GEMM operand carries x_mask
            }
        }
        __syncthreads();

        // ---- Phase 2: WMMA over this K chunk ----
        const float* a0p = xb + (size_t)(cBase      + lane16) * T_ + kb + 2 * hi;
        const float* a1p = xb + (size_t)(cBase + 16 + lane16) * T_ + kb + 2 * hi;
        const float* bp  = &Elds[lane16 * ESTRIDE + 2 * hi];

        #pragma unroll 2
        for (int k0 = 0; k0 < KCH; k0 += 4) {
            const v2f a0 = *(const v2f*)(a0p + k0);
            const v2f a1 = *(const v2f*)(a1p + k0);
            const v2f b0 = *(const v2f*)(bp + 0 * 16 * ESTRIDE + k0);
            const v2f b1 = *(const v2f*)(bp + 1 * 16 * ESTRIDE + k0);
            const v2f b2 = *(const v2f*)(bp + 2 * 16 * ESTRIDE + k0);
            const v2f b3 = *(const v2f*)(bp + 3 * 16 * ESTRIDE + k0);

            acc[0][0] = __builtin_amdgcn_wmma_f32_16x16x4_f32(false, a0, false, b0, (short)0, acc[0][0], false, false);
            acc[0][1] = __builtin_amdgcn_wmma_f32_16x16x4_f32(false, a0, false, b1, (short)0, acc[0][1], false, false);
            acc[0][2] = __builtin_amdgcn_wmma_f32_16x16x4_f32(false, a0, false, b2, (short)0, acc[0][2], false, false);
            acc[0][3] = __builtin_amdgcn_wmma_f32_16x16x4_f32(false, a0, false, b3, (short)0, acc[0][3], false, false);
            acc[1][0] = __builtin_amdgcn_wmma_f32_16x16x4_f32(false, a1, false, b0, (short)0, acc[1][0], false, false);
            acc[1][1] = __builtin_amdgcn_wmma_f32_16x16x4_f32(false, a1, false, b1, (short)0, acc[1][1], false, false);
            acc[1][2] = __builtin_amdgcn_wmma_f32_16x16x4_f32(false, a1, false, b2, (short)0, acc[1][2], false, false);
            acc[1][3] = __builtin_amdgcn_wmma_f32_16x16x4_f32(false, a1, false, b3, (short)0, acc[1][3], false, false);
        }
        __syncthreads();  // before next chunk overwrites Elds
    }

    // ---- normalizer reduction: denom[l] = sum_t E, inv = ymask/(denom+eps) ----
    s_part[kseg * 64 + n] = psum;
    __syncthreads();
    if (tid < 64) {
        const float tot = s_part[tid] + s_part[64 + tid] + s_part[128 + tid] + s_part[192 + tid];
        s_inv[tid] = ymask[(size_t)b * L_ + lBase + tid] / (tot + EPSF);
    }
    __syncthreads();

    // ---- epilogue: scale columns and store ----
    // C/D layout: VGPR r holds M=r (lanes 0-15) / M=r+8 (lanes 16-31), N=lane%16
    #pragma unroll
    for (int mt = 0; mt < 2; ++mt) {
        const int c0 = cBase + mt * 16 + 8 * hi;
        #pragma unroll
        for (int nt = 0; nt < 4; ++nt) {
            const float sc = s_inv[nt * 16 + lane16];
            const int lcol = lBase + nt * 16 + lane16;
            float* po = out + ((size_t)b * C_ + c0) * L_ + lcol;
            const v8f a = acc[mt][nt];
            #pragma unroll
            for (int r = 0; r < 8; ++r)
                po[(size_t)r * L_] = a[r] * sc;
        }
    }
}

extern "C" void kernel_launch(void* const* d_in, const int* in_sizes, int n_in,
                              void* d_out, int out_size, void* d_ws, size_t ws_size,
                              hipStream_t stream) {
    (void)in_sizes; (void)n_in; (void)out_size; (void)ws_size;
    const float* x  = (const float*)d_in[0];  // (16,256,512)
    const float* w  = (const float*)d_in[1];  // (16,512)
    const float* xm = (const float*)d_in[2];  // (16,1,512)
    const float* ym = (const float*)d_in[3];  // (16,1,4096)
    const float* ss = (const float*)d_in[4];  // (1,)
    float* out = (float*)d_out;               // (16,256,4096)

    float* center = (float*)d_ws;             // 16*512 f32
    float* coef   = center + B_ * T_;         // 16*512 f32  (64 KB total)

    dlr_scan_kernel<<<B_, T_, 0, stream>>>(w, ss, center, coef);
    dlr_upsample_kernel<<<B_ * 64, 256, 0, stream>>>(x, xm, ym, center, coef, out);
}